// AlignLoss3_5506148073593
// MI455X (gfx1250) — compile-verified
//
#include <hip/hip_runtime.h>
#include <hip/hip_bf16.h>
#include <math.h>

typedef __attribute__((ext_vector_type(2))) float v2f;
typedef __attribute__((ext_vector_type(8))) float v8f;

#define TAU   0.07f
#define EPSN  1e-12f
#define NROW  8192
#define DDIM  512
#define NCLS  7

// ---------------------------------------------------------------------------
// Kernel 1: normalize anchors into zero-padded 16 x 512 buffer; zero counts.
// ---------------------------------------------------------------------------
__global__ __launch_bounds__(256) void prep_kernel(const float* __restrict__ anchors,
                                                   float* __restrict__ normA,
                                                   int* __restrict__ counts) {
    __shared__ float red[256];
    const int t = threadIdx.x;
    if (t < 16) counts[t] = 0;
    // zero pad rows 7..15
    for (int idx = t; idx < (16 - NCLS) * DDIM; idx += 256) normA[NCLS * DDIM + idx] = 0.0f;

    for (int c = 0; c < NCLS; ++c) {
        float s = 0.0f;
        for (int k = t; k < DDIM; k += 256) { float v = anchors[c * DDIM + k]; s += v * v; }
        red[t] = s;
        __syncthreads();
        for (int off = 128; off > 0; off >>= 1) {
            if (t < off) red[t] += red[t + off];
            __syncthreads();
        }
        const float inv = 1.0f / fmaxf(sqrtf(red[0]), EPSN);
        __syncthreads();
        for (int k = t; k < DDIM; k += 256) normA[c * DDIM + k] = anchors[c * DDIM + k] * inv;
        __syncthreads();
    }
}

// ---------------------------------------------------------------------------
// Kernel 2: per-row argmax of Y (first-max tie-break) + class histogram.
// ---------------------------------------------------------------------------
__global__ __launch_bounds__(256) void label_kernel(const float* __restrict__ Y,
                                                    int* __restrict__ labels,
                                                    int* __restrict__ counts) {
    const int i = blockIdx.x * 256 + threadIdx.x;
    if (i >= NROW) return;
    const float* y = Y + (size_t)i * NCLS;
    float best = y[0];
    int   bl   = 0;
#pragma unroll
    for (int c = 1; c < NCLS; ++c) {
        float v = y[c];
        if (v > best) { best = v; bl = c; }
    }
    labels[i] = bl;
    atomicAdd(&counts[bl], 1);
}

// ---------------------------------------------------------------------------
// Kernel 3: S = X @ normA^T via V_WMMA_F32_16X16X4_F32, then per-row
// count-weighted logsumexp. One wave per 16-row tile; 8 waves per block.
// ---------------------------------------------------------------------------
__global__ __launch_bounds__(256) void gemm_lse_kernel(const float* __restrict__ X,
                                                       const float* __restrict__ normA,
                                                       const int* __restrict__ labels,
                                                       const int* __restrict__ counts,
                                                       float* __restrict__ rowLoss) {
    __shared__ float sB[16 * DDIM];     // padded normalized anchors (32 KB)
    __shared__ float sS[8][16][17];     // per-wave 16x16 S tiles (pad vs bank conflicts)
    __shared__ float sN[8][16];         // per-wave row norms

    // Cooperative, coalesced load of padded anchors into LDS.
    for (int idx = threadIdx.x; idx < 16 * DDIM; idx += 256) sB[idx] = normA[idx];
    __syncthreads();

    const int wave  = threadIdx.x >> 5;
    const int lane  = threadIdx.x & 31;
    const int m     = lane & 15;          // row within tile (A) / col within tile (B)
    const int kbase = (lane >> 4) << 1;   // lanes 0-15 -> K {0,1}; lanes 16-31 -> K {2,3}
    const int i0    = blockIdx.x * 128 + wave * 16;

    const float* xrow = X + (size_t)(i0 + m) * DDIM;

    v8f   acc   = {};
    float sumsq = 0.0f;
    for (int kk = 0; kk < DDIM; kk += 4) {
        v2f a = *(const v2f*)(xrow + kk + kbase);           // A: 16x4 f32 tile of X
        v2f b = *(const v2f*)(&sB[m * DDIM + kk + kbase]);  // B: 4x16 f32 tile of normA^T
        sumsq += a.x * a.x + a.y * a.y;                     // co-executes on VALU
        acc = __builtin_amdgcn_wmma_f32_16x16x4_f32(
            /*neg_a=*/false, a, /*neg_b=*/false, b,
            /*c_mod=*/(short)0, acc, /*reuse_a=*/false, /*reuse_b=*/false);
    }
    // lanes l and l+16 each summed half of row m's squares -> combine (wave32)
    sumsq += __shfl_xor(sumsq, 16, 32);

    // Stage C tile: VGPR r of lane l holds S[r + 8*(l>>4)][l&15]
    const int hi = lane >> 4;
#pragma unroll
    for (int r = 0; r < 8; ++r) sS[wave][r + 8 * hi][m] = acc[r];
    if (lane < 16) sN[wave][m] = sqrtf(sumsq);
    __syncthreads();

    if (lane < 16) {
        const int   row  = i0 + m;
        const float nrm  = fmaxf(sN[wave][m], EPSN);
        const float invs = 1.0f / (nrm * TAU);

        float z[NCLS];
        int   cnt[NCLS];
        float mx = -INFINITY;
#pragma unroll
        for (int c = 0; c < NCLS; ++c) {
            cnt[c] = counts[c];
            z[c]   = sS[wave][m][c] * invs;
            if (cnt[c] > 0 && z[c] > mx) mx = z[c];
        }
        float sum = 0.0f;
#pragma unroll
        for (int c = 0; c < NCLS; ++c)
            if (cnt[c] > 0) sum += (float)cnt[c] * expf(z[c] - mx);

        const float lse = mx + logf(sum);
        rowLoss[row] = lse - z[labels[row]];
    }
}

// ---------------------------------------------------------------------------
// Kernel 4: deterministic fixed-order reduction of per-row losses -> mean.
// ---------------------------------------------------------------------------
__global__ __launch_bounds__(256) void reduce_kernel(const float* __restrict__ rowLoss,
                                                     float* __restrict__ out) {
    __shared__ float red[256];
    const int t = threadIdx.x;
    float s = 0.0f;
    for (int i = t; i < NROW; i += 256) s += rowLoss[i];
    red[t] = s;
    __syncthreads();
    for (int off = 128; off > 0; off >>= 1) {
        if (t < off) red[t] += red[t + off];
        __syncthreads();
    }
    if (t == 0) out[0] = red[0] * (1.0f / (float)NROW);  // WEIGHT = 1.0
}

// ---------------------------------------------------------------------------
extern "C" void kernel_launch(void* const* d_in, const int* in_sizes, int n_in,
                              void* d_out, int out_size, void* d_ws, size_t ws_size,
                              hipStream_t stream) {
    const float* X       = (const float*)d_in[0];  // (8192, 512)
    const float* Y       = (const float*)d_in[1];  // (8192, 7)
    const float* anchors = (const float*)d_in[2];  // (7, 512)

    float* normA   = (float*)d_ws;           // 16*512 floats (zero-padded)
    int*   labels  = (int*)(normA + 16 * DDIM);  // 8192 ints
    int*   counts  = labels + NROW;              // 16 ints
    float* rowLoss = (float*)(counts + 16);      // 8192 floats

    prep_kernel<<<1, 256, 0, stream>>>(anchors, normA, counts);
    label_kernel<<<NROW / 256, 256, 0, stream>>>(Y, labels, counts);
    gemm_lse_kernel<<<NROW / 128, 256, 0, stream>>>(X, normA, labels, counts, rowLoss);
    reduce_kernel<<<1, 256, 0, stream>>>(rowLoss, (float*)d_out);
}